// Elastodynamics_28810640622190
// MI455X (gfx1250) — compile-verified
//
#include <hip/hip_runtime.h>
#include <math.h>

// Second-order forward-mode PINN evaluation:
//   per point carry 10 channels (value, 3 jac dirs, 6 sym Hessian entries)
//   of a 50-wide feature vector through 6 tanh layers.
// Linear part of every layer = 10-channel batched matvec -> f32 WMMA
// (V_WMMA_F32_16X16X4_F32, the only f32 WMMA shape on CDNA5).
// Activation uses the CDNA5 hardware V_TANH_F32 when the builtin exists.

typedef __attribute__((ext_vector_type(2))) float v2f;
typedef __attribute__((ext_vector_type(8))) float v8f;

#if defined(__has_builtin)
#  if __has_builtin(__builtin_amdgcn_tanhf)
#    define TANHF(x) __builtin_amdgcn_tanhf(x)
#  elif __has_builtin(__builtin_amdgcn_tanh_f32)
#    define TANHF(x) __builtin_amdgcn_tanh_f32(x)
#  else
#    define TANHF(x) tanhf(x)
#  endif
#else
#  define TANHF(x) tanhf(x)
#endif

#define WIDTH   50
#define NHID    5
#define RHO_C   0.284f
#define BODYFX  1.86f
#define NCH     10     // value, j0,j1,j2, h00,h01,h02,h11,h12,h22
#define PTS     16     // points per wave (M of WMMA tile)
#define FPAD    64     // feature dim padded to 4 tiles of 16
#define KPAD    52     // K dim padded to 13 chunks of 4

__launch_bounds__(32, 1)
__global__ void elasto_pinn_kernel(const float* __restrict__ x,
                                   const float* __restrict__ y,
                                   const float* __restrict__ t,
                                   const float* __restrict__ W_in,
                                   const float* __restrict__ b_in,
                                   const float* __restrict__ W_h,
                                   const float* __restrict__ b_h,
                                   const float* __restrict__ W_out,
                                   const float* __restrict__ b_out,
                                   const float* __restrict__ cp11,
                                   const float* __restrict__ cp12,
                                   const float* __restrict__ cp22,
                                   const float* __restrict__ cp33,
                                   float* __restrict__ out, int n)
{
    __shared__ float hbuf[NCH][PTS][FPAD];   // state, point-major (A-source)
    __shared__ float WT[FPAD][KPAD];         // WT[outCol][inRow], zero-padded
    __shared__ float bias_s[FPAD];

    const int lane = threadIdx.x;            // wave32: 0..31
    const int pt0  = blockIdx.x * PTS;

    // ---- zero state, then seed input channels ------------------------------
    for (int i = lane; i < NCH * PTS * FPAD; i += 32)
        ((float*)hbuf)[i] = 0.0f;
    __syncthreads();
    if (lane < PTS) {
        int g = pt0 + lane;
        if (g < n) {
            hbuf[0][lane][0] = x[g];         // value channel = (x,y,t)
            hbuf[0][lane][1] = y[g];
            hbuf[0][lane][2] = t[g];
            hbuf[1][lane][0] = 1.0f;         // jac channels = unit tangents
            hbuf[2][lane][1] = 1.0f;
            hbuf[3][lane][2] = 1.0f;         // hess channels stay 0
        }
    }
    __syncthreads();

    const int pt    = lane & 15;             // A: point row / B,C: column
    const int khalf = (lane >> 4) << 1;      // lanes 0-15 -> K{0,1}; 16-31 -> K{2,3}
    const int prow  = (lane >> 4) * 8;       // C layout: vgpr r -> M = r (+8)
    const v8f  vzero = {0.f,0.f,0.f,0.f,0.f,0.f,0.f,0.f};

    for (int layer = 0; layer <= NHID; ++layer) {
        const int   Kin  = (layer == 0) ? 3 : WIDTH;
        const int   nkc  = (layer == 0) ? 1 : (KPAD / 4);
        const float* Ws  = (layer == 0) ? W_in : (W_h + (size_t)(layer-1)*WIDTH*WIDTH);
        const float* bs  = (layer == 0) ? b_in : (b_h + (size_t)(layer-1)*WIDTH);

        // stage W transposed + padded with exact zeros (keeps pad lanes at 0)
        for (int c = lane; c < FPAD; c += 32) {
            #pragma unroll 4
            for (int r = 0; r < KPAD; ++r)
                WT[c][r] = (c < WIDTH && r < Kin) ? Ws[r * WIDTH + c] : 0.0f;
        }
        for (int i = lane; i < FPAD; i += 32)
            bias_s[i] = (i < WIDTH) ? bs[i] : 0.0f;
        __syncthreads();

        // prefetch next layer's weights into cache while we compute this one
        if (layer < NHID) {
            const float* Wnext = W_h + (size_t)layer * WIDTH * WIDTH;
            __builtin_prefetch(Wnext + lane * 64, 0, 3);   // ~cacheline strided
            __builtin_prefetch(Wnext + 2048 + lane * 16, 0, 3);
        }

        // ---- z_c = h_c @ W via f32 WMMA -----------------------------------
        v8f acc[NCH][4];
        #pragma unroll
        for (int c = 0; c < NCH; ++c)
            #pragma unroll
            for (int nt = 0; nt < 4; ++nt)
                acc[c][nt] = vzero;

        for (int kc = 0; kc < nkc; ++kc) {
            const int k = kc * 4 + khalf;    // even -> 8B-aligned b64 LDS loads
            v2f a[NCH];
            #pragma unroll
            for (int c = 0; c < NCH; ++c)
                a[c] = *(const v2f*)&hbuf[c][pt][k];
            v2f b[4];
            #pragma unroll
            for (int nt = 0; nt < 4; ++nt)
                b[nt] = *(const v2f*)&WT[pt + nt * 16][k];
            #pragma unroll
            for (int c = 0; c < NCH; ++c)
                #pragma unroll
                for (int nt = 0; nt < 4; ++nt)
                    acc[c][nt] = __builtin_amdgcn_wmma_f32_16x16x4_f32(
                        false, a[c], false, b[nt], (short)0, acc[c][nt],
                        false, false);
        }
        __syncthreads();                      // A reads done before overwrite

        // ---- tanh chain rule, elementwise across the 10 accumulators ------
        #pragma unroll
        for (int nt = 0; nt < 4; ++nt) {
            const float bc = bias_s[pt + nt * 16];
            #pragma unroll
            for (int e = 0; e < 8; ++e) {
                float z   = acc[0][nt][e] + bc;
                float h   = TANHF(z);
                float tp  = 1.0f - h * h;
                float tpp = -2.0f * h * tp;
                float j0 = acc[1][nt][e], j1 = acc[2][nt][e], j2 = acc[3][nt][e];
                acc[4][nt][e] = tpp * j0 * j0 + tp * acc[4][nt][e];
                acc[5][nt][e] = tpp * j0 * j1 + tp * acc[5][nt][e];
                acc[6][nt][e] = tpp * j0 * j2 + tp * acc[6][nt][e];
                acc[7][nt][e] = tpp * j1 * j1 + tp * acc[7][nt][e];
                acc[8][nt][e] = tpp * j1 * j2 + tp * acc[8][nt][e];
                acc[9][nt][e] = tpp * j2 * j2 + tp * acc[9][nt][e];
                acc[1][nt][e] = tp * j0;
                acc[2][nt][e] = tp * j1;
                acc[3][nt][e] = tp * j2;
                acc[0][nt][e] = h;
            }
        }

        // ---- write C-layout tiles back into point-major state -------------
        #pragma unroll
        for (int c = 0; c < NCH; ++c)
            #pragma unroll
            for (int nt = 0; nt < 4; ++nt)
                #pragma unroll
                for (int e = 0; e < 8; ++e)
                    hbuf[c][prow + e][pt + nt * 16] = acc[c][nt][e];
        __syncthreads();
    }

    // ---- output layer (50 -> 2) + Hessian recombination, scalar per point --
    if (lane < PTS) {
        int g = pt0 + lane;
        if (g < n) {
            float sv0=0.f, sv1=0.f, s00_0=0.f, s00_1=0.f, s01_0=0.f, s01_1=0.f,
                  s11_0=0.f, s11_1=0.f, s22_0=0.f, s22_1=0.f;
            for (int f = 0; f < WIDTH; ++f) {
                float w0 = W_out[f*2+0], w1 = W_out[f*2+1];
                float hv  = hbuf[0][lane][f];
                float h00 = hbuf[4][lane][f];
                float h01 = hbuf[5][lane][f];
                float h11 = hbuf[7][lane][f];
                float h22 = hbuf[9][lane][f];
                sv0  += hv  * w0;  sv1  += hv  * w1;
                s00_0 += h00 * w0; s00_1 += h00 * w1;
                s01_0 += h01 * w0; s01_1 += h01 * w1;
                s11_0 += h11 * w0; s11_1 += h11 * w1;
                s22_0 += h22 * w0; s22_1 += h22 * w1;
            }
            float c11 = cp11[0], c12 = cp12[0], c22 = cp22[0], c33 = cp33[0];
            float mx = sv0 + b_out[0];
            float my = sv1 + b_out[1];
            float ax = s22_0, ay = s22_1;
            float lqx  = c11 * s00_0 + c12 * s01_1;   // c11*H00@w0 + c12*H01@w1
            float lqxy = c33 * (s01_0 + s00_1);       // c33*(H01@w0 + H00@w1)
            float lqyx = c33 * (s11_0 + s01_1);       // c33*(H11@w0 + H01@w1)
            float lqyy = c12 * s01_0 + c22 * s11_1;   // c12*H01@w0 + c22*H11@w1
            float fx = RHO_C * ax - lqx - lqyx - BODYFX;
            float fy = RHO_C * ay - lqxy - lqyy;
            out[g]         = mx;
            out[n + g]     = my;
            out[2*n + g]   = fx;
            out[3*n + g]   = fy;
        }
    }
}

extern "C" void kernel_launch(void* const* d_in, const int* in_sizes, int n_in,
                              void* d_out, int out_size, void* d_ws, size_t ws_size,
                              hipStream_t stream)
{
    const float* x     = (const float*)d_in[0];
    const float* y     = (const float*)d_in[1];
    const float* t     = (const float*)d_in[2];
    const float* W_in  = (const float*)d_in[3];
    const float* b_in  = (const float*)d_in[4];
    const float* W_h   = (const float*)d_in[5];
    const float* b_h   = (const float*)d_in[6];
    const float* W_out = (const float*)d_in[7];
    const float* b_out = (const float*)d_in[8];
    const float* c11   = (const float*)d_in[9];
    const float* c12   = (const float*)d_in[10];
    const float* c22   = (const float*)d_in[11];
    const float* c33   = (const float*)d_in[12];
    float* out = (float*)d_out;

    const int n = in_sizes[0];
    const int blocks = (n + PTS - 1) / PTS;   // one wave32 per block, 16 pts/wave
    elasto_pinn_kernel<<<blocks, 32, 0, stream>>>(
        x, y, t, W_in, b_in, W_h, b_h, W_out, b_out,
        c11, c12, c22, c33, out, n);
}